// AdaptiveSSDLoss_43679817400828
// MI455X (gfx1250) — compile-verified
//
#include <hip/hip_runtime.h>
#include <cstdint>

#define N_ 64
#define C_ 81
#define A_ 8732

typedef float v2f __attribute__((ext_vector_type(2)));
typedef float v8f __attribute__((ext_vector_type(8)));

// ---------------------------------------------------------------------------
// Kernel A: focal loss per anchor.  con[n,a] = -(1-pt)^2 * log(pt)
// Streams plabel (181 MB) once with non-temporal loads.  Register-chunked
// softmax (3 chunks of 27 classes) => ~1.07 exp per element instead of 2.
// ---------------------------------------------------------------------------
__global__ __launch_bounds__(256) void focal_kernel(
    const float* __restrict__ plabel, const int* __restrict__ glabel,
    float* __restrict__ con) {
  const int n  = blockIdx.y;
  const int a0 = blockIdx.x * 256 + threadIdx.x;
  const bool valid = a0 < A_;
  const int a = valid ? a0 : (A_ - 1);
  const int g = glabel[n * A_ + a];
  const float* __restrict__ base = plabel + (size_t)n * (C_ * A_) + a;

  float m  = -__builtin_inff();
  float s  = 0.f;
  float xg = 0.f;
#pragma unroll
  for (int chunk = 0; chunk < 3; ++chunk) {
    float v[27];
    float cm = -__builtin_inff();
#pragma unroll
    for (int j = 0; j < 27; ++j) {
      const int c = chunk * 27 + j;
      v[j] = __builtin_nontemporal_load(base + (size_t)c * A_);
      cm = fmaxf(cm, v[j]);
      xg = (c == g) ? v[j] : xg;
    }
    float cs = 0.f;
#pragma unroll
    for (int j = 0; j < 27; ++j) cs += __expf(v[j] - cm);
    const float nm = fmaxf(m, cm);
    s = s * __expf(m - nm) + cs * __expf(cm - nm);
    m = nm;
  }
  const float logpt = xg - (m + __logf(s));
  const float pt = __expf(logpt);
  const float om = 1.f - pt;
  float c_out = -(om * om) * logpt;
  c_out = fmaxf(c_out, 0.f);  // kill -0.0 so uint-ordered radix select is valid
  if (valid) con[(size_t)n * A_ + a0] = c_out;
}

// ---------------------------------------------------------------------------
// Block reduction helper (1024 threads)
// ---------------------------------------------------------------------------
__device__ __forceinline__ float block_reduce(float v, float* s_red, int tid) {
  s_red[tid] = v;
  __syncthreads();
  for (int off = 512; off > 0; off >>= 1) {
    if (tid < off) s_red[tid] += s_red[tid + off];
    __syncthreads();
  }
  float r = s_red[0];
  __syncthreads();
  return r;
}

__device__ __forceinline__ unsigned lds_off_of(const void* p) {
  // generic -> LDS address space; the as(3) pointer value IS the LDS byte
  // offset relative to the workgroup's LDS base (what the async op wants).
  return (unsigned)(unsigned long long)(__attribute__((address_space(3))) const char*)p;
}

__device__ __forceinline__ float smooth_l1(float x) {
  float ax = fabsf(x);
  return ax < 1.f ? 0.5f * x * x : ax - 0.5f;
}

// ---------------------------------------------------------------------------
// Kernel B: one block per batch row.  con row is async-copied into LDS
// (global_load_async_to_lds_b128, waited with s_wait_asynccnt) and all
// selection passes run out of LDS:
//   - pos count, smooth-L1 (source domain), positive-confidence sum
//   - 4-pass radix select of the K-th largest con_neg  (K = min(3*pos, A))
//   - sum of strictly-greater values + exact tie handling:
//       pivot > 0 : ties all equal pivot -> rem * pivot
//       pivot == 0: stable index-ordered selection of first `rem` zero-key
//                   anchors, summing FULL con (positives contribute!)
// ---------------------------------------------------------------------------
__global__ __launch_bounds__(1024) void row_kernel(
    const float* __restrict__ ploc, const float* __restrict__ gloc,
    const int* __restrict__ glabel, const int* __restrict__ domain,
    const float* __restrict__ con, float* __restrict__ row_loss) {
  const int n = blockIdx.x;
  const int tid = threadIdx.x;

  __shared__ float s_con[A_];            // 34928 B
  __shared__ unsigned char s_pos[A_];    //  8732 B
  __shared__ unsigned s_bins[256];
  __shared__ float s_red[1024];
  __shared__ unsigned s_wcnt[32];
  __shared__ unsigned s_prefix, s_rem, s_base;

  // ---- async copy of this row's con[] into LDS (ASYNCcnt path) ----
  {
    const char* gsrc = (const char*)(con + (size_t)n * A_);
    const unsigned lds_base = lds_off_of(&s_con[0]);
    for (int i = tid; i < (A_ * 4) / 16; i += 1024) {   // 2183 x b128 chunks
      unsigned ldso = lds_base + (unsigned)(i * 16);
      unsigned long long ga = (unsigned long long)(gsrc + (size_t)i * 16);
      asm volatile("global_load_async_to_lds_b128 %0, %1, off"
                   :: "v"(ldso), "v"(ga) : "memory");
    }
  }

  // overlap: build positive mask + count while the async copies fly
  float cntf = 0.f;
  for (int a = tid; a < A_; a += 1024) {
    const unsigned char p = (glabel[n * A_ + a] > 0) ? 1 : 0;
    s_pos[a] = p;
    cntf += (float)p;
  }
  asm volatile("s_wait_asynccnt 0" ::: "memory");
  __syncthreads();

  // ---- smooth-L1 + positive-confidence (source domain only) ----
  const bool src = (domain[n] == 0);
  float sl1 = 0.f, pconf = 0.f;
  if (src) {
    for (int a = tid; a < A_; a += 1024) {
      if (s_pos[a]) {
        const float4 gl = *(const float4*)(gloc + ((size_t)n * A_ + a) * 4);
        const float p0 = ploc[((size_t)n * 4 + 0) * A_ + a];
        const float p1 = ploc[((size_t)n * 4 + 1) * A_ + a];
        const float p2 = ploc[((size_t)n * 4 + 2) * A_ + a];
        const float p3 = ploc[((size_t)n * 4 + 3) * A_ + a];
        sl1 += smooth_l1(p0 - gl.x) + smooth_l1(p1 - gl.y) +
               smooth_l1(p2 - gl.z) + smooth_l1(p3 - gl.w);
        pconf += s_con[a];
      }
    }
  }
  cntf  = block_reduce(cntf, s_red, tid);
  sl1   = block_reduce(sl1, s_red, tid);
  pconf = block_reduce(pconf, s_red, tid);

  const int pos_num = (int)(cntf + 0.5f);
  int K = 3 * pos_num;
  if (K > A_) K = A_;

  float sum_neg = 0.f;
  if (K > 0) {
    // ---- 4-pass MSB-first radix select of K-th largest key over LDS ----
    if (tid == 0) { s_prefix = 0u; s_rem = (unsigned)K; }
    __syncthreads();
    for (int pass = 0; pass < 4; ++pass) {
      const int shift = 24 - 8 * pass;
      if (tid < 256) s_bins[tid] = 0u;
      __syncthreads();
      const unsigned prefix = s_prefix;
      const unsigned himask = (pass == 0) ? 0u : (0xFFFFFFFFu << (shift + 8));
      for (int a = tid; a < A_; a += 1024) {
        const float v = s_pos[a] ? 0.f : s_con[a];
        const unsigned key = __float_as_uint(v);
        if ((key & himask) == prefix)
          atomicAdd(&s_bins[(key >> shift) & 255u], 1u);
      }
      __syncthreads();
      if (tid == 0) {
        unsigned cum = 0, rem = s_rem, sel = 0;
        for (int b = 255; b >= 0; --b) {
          const unsigned c = s_bins[b];
          if (cum + c >= rem) { sel = (unsigned)b; break; }
          cum += c;
        }
        s_prefix = prefix | (sel << shift);
        s_rem = rem - cum;
      }
      __syncthreads();
    }
    const unsigned P = s_prefix;
    const unsigned rem = s_rem;
    const float pv = __uint_as_float(P);

    // strictly-greater-than-pivot sum (all such entries are negatives)
    float sgt = 0.f;
    for (int a = tid; a < A_; a += 1024) {
      const float v = s_pos[a] ? 0.f : s_con[a];
      if (__float_as_uint(v) > P) sgt += v;
    }
    sum_neg = block_reduce(sgt, s_red, tid);

    if (P != 0u) {
      sum_neg += (float)rem * pv;   // tied entries all contribute pivot value
    } else {
      // stable (index-ordered) selection of first `rem` zero-key anchors,
      // summing FULL con (positives have con_neg==0 but con!=0).
      float acc = 0.f;
      if (tid == 0) s_base = 0u;
      __syncthreads();
      const int wid = tid >> 5, lane = tid & 31;  // wave32
      for (int start = 0; start < A_; start += 1024) {
        const int a = start + tid;
        const bool flag =
            (a < A_) && (s_pos[a] ? true : (s_con[a] == 0.f));
        const unsigned long long bal = __ballot(flag);
        if (lane == 0) s_wcnt[wid] = (unsigned)__popcll(bal);
        __syncthreads();
        unsigned woff = 0;
        for (int w = 0; w < wid; ++w) woff += s_wcnt[w];
        const unsigned rank =
            s_base + woff + (unsigned)__popcll(bal & ((1ull << lane) - 1ull));
        if (flag && rank < rem) acc += s_con[a];
        __syncthreads();
        if (tid == 0) {
          unsigned tot = 0;
          for (int w = 0; w < 32; ++w) tot += s_wcnt[w];
          s_base += tot;
        }
        __syncthreads();
      }
      sum_neg += block_reduce(acc, s_red, tid);
    }
  }

  if (tid == 0) {
    const float total = sl1 + pconf + sum_neg;  // ALPHA = 1
    row_loss[n] =
        (pos_num > 0) ? total / fmaxf((float)pos_num, 1e-6f) : 0.f;
  }
}

// ---------------------------------------------------------------------------
// Kernel C: mean over the 64 rows via v_wmma_f32_16x16x4_f32.
//
// With A = all-ones(16x4):  D[m][n] = sum_k B[k][n]  (+C=0).  Summing one D
// register across all lanes therefore yields  2 * sum(all 64 B slots)
// (each of the 16 identical D rows appears twice across the 32 lanes of
// d[0]).  This is correct for ANY bijective lane->(k,n) layout of B, so the
// reduction is immune to B-layout details.  Single full wave => EXEC all 1s.
// ---------------------------------------------------------------------------
__global__ __launch_bounds__(32) void finalize_kernel(
    const float* __restrict__ row_loss, float* __restrict__ out) {
  const int lane = threadIdx.x;            // 0..31, one wave32
  v2f a = {1.f, 1.f};                      // A = ones (any layout)
  v2f b = {row_loss[lane], row_loss[lane + 32]};  // all 64 values, bijective
  v8f c = {};
  // 8 args: (neg_a, A, neg_b, B, c_mod, C, reuse_a, reuse_b)
  v8f d = __builtin_amdgcn_wmma_f32_16x16x4_f32(false, a, false, b, (short)0,
                                                c, false, false);
  float t = d[0];
#pragma unroll
  for (int off = 16; off > 0; off >>= 1) t += __shfl_xor(t, off, 32);
  // t = 2 * sum(row_loss);  mean = t / (2*64)
  if (lane == 0) out[0] = t * (1.f / 128.f);
}

// ---------------------------------------------------------------------------
extern "C" void kernel_launch(void* const* d_in, const int* in_sizes, int n_in,
                              void* d_out, int out_size, void* d_ws,
                              size_t ws_size, hipStream_t stream) {
  const float* ploc   = (const float*)d_in[0];  // [N,4,A]
  const float* plabel = (const float*)d_in[1];  // [N,C,A]
  const float* gloc   = (const float*)d_in[2];  // [N,A,4]
  const int*   glabel = (const int*)d_in[3];    // [N,A]
  const int*   domain = (const int*)d_in[4];    // [N]
  float* out = (float*)d_out;

  float* con      = (float*)d_ws;               // N*A floats
  float* row_loss = con + (size_t)N_ * A_;      // N floats

  dim3 gridA((A_ + 255) / 256, N_);
  focal_kernel<<<gridA, 256, 0, stream>>>(plabel, glabel, con);
  row_kernel<<<N_, 1024, 0, stream>>>(ploc, gloc, glabel, domain, con,
                                      row_loss);
  finalize_kernel<<<1, 32, 0, stream>>>(row_loss, out);
}